// my_CorrCriterion_16913581211755
// MI455X (gfx1250) — compile-verified
//
#include <hip/hip_runtime.h>
#include <math.h>

typedef float v8f __attribute__((ext_vector_type(8)));
typedef float v2f __attribute__((ext_vector_type(2)));

#define B_SZ 2
#define N_PTS 8192
#define M_TOTAL (B_SZ * N_PTS)

#define CHUNK_COLS 2048                      // columns staged in LDS per block iteration
#define NUM_CHUNKS (M_TOTAL / CHUNK_COLS)    // 8
#define CHUNK_TILES (CHUNK_COLS / 16)        // 128

// ws accumulators: [0]=sum(w*err) [1]=sum(w) [2]=sum_bce_label1 [3]=cnt1 [4]=sum_bce_label0 [5]=cnt0
__global__ void cc_init_kernel(float* __restrict__ acc) {
    if (threadIdx.x < 6) acc[threadIdx.x] = 0.0f;
}

// Compute gt = R*kp + t, pack {g0,g1,g2,|g|^2} as float4; accumulate weighted L1 corr error.
__global__ void cc_prep_kernel(const float* __restrict__ kp_before,
                               const float* __restrict__ pred,
                               const float* __restrict__ pose,
                               const float* __restrict__ w,
                               float4* __restrict__ gt4,
                               float* __restrict__ acc) {
    int i = blockIdx.x * blockDim.x + threadIdx.x;
    if (i >= M_TOTAL) return;
    int b = i / N_PTS;
    const float* P = pose + b * 12;            // pose_gt[b] is 3x4 row-major
    float x = kp_before[i * 3 + 0];
    float y = kp_before[i * 3 + 1];
    float z = kp_before[i * 3 + 2];
    float g0 = P[0] * x + P[1] * y + P[2]  * z + P[3];
    float g1 = P[4] * x + P[5] * y + P[6]  * z + P[7];
    float g2 = P[8] * x + P[9] * y + P[10] * z + P[11];
    float gn = g0 * g0 + g1 * g1 + g2 * g2;
    gt4[i] = make_float4(g0, g1, g2, gn);
    float p0 = pred[i * 3 + 0], p1 = pred[i * 3 + 1], p2 = pred[i * 3 + 2];
    float err = fabsf(p0 - g0) + fabsf(p1 - g1) + fabsf(p2 - g2);
    float wi = w[i];
    atomicAdd(&acc[0], wi * err);
    atomicAdd(&acc[1], wi);
}

// Fused pairwise-distance argmin via V_WMMA_F32_16X16X4_F32 with LDS-staged columns.
// A (16x4) = [p0, p1, p2, |p|^2] ; B (4x16) = [-2g0, -2g1, -2g2, 1]^T per column
//   => C[m][n] = -2*p_m.g_n + |p_m|^2 ; sq = C + |g_n|^2
// One wave per 16-row tile; the 8 waves of a block share LDS column chunks.
__global__ void __launch_bounds__(256)
cc_argmin_kernel(const float* __restrict__ pred,
                 const float4* __restrict__ gt4,
                 const float* __restrict__ logits,
                 float* __restrict__ acc) {
    __shared__ float4 sh[CHUNK_COLS];          // 32 KB of the 320 KB WGP LDS

    const int tid  = threadIdx.x;
    const int lane = tid & 31;
    const int wid  = blockIdx.x * (blockDim.x >> 5) + (tid >> 5);
    const int rowbase = wid * 16;
    const int c16  = lane & 15;
    const bool hiHalf = (lane >= 16);

    // A-matrix per ISA layout: lanes 0-15 hold (K=0,K=1), lanes 16-31 hold (K=2,K=3), M = lane&15
    const int m = rowbase + c16;
    float p0 = pred[m * 3 + 0];
    float p1 = pred[m * 3 + 1];
    float p2 = pred[m * 3 + 2];
    float pn = p0 * p0 + p1 * p1 + p2 * p2;
    v2f a;
    a.x = hiHalf ? p2 : p0;
    a.y = hiHalf ? pn : p1;

    float    minv[8];
    unsigned mini[8];
#pragma unroll
    for (int r = 0; r < 8; ++r) { minv[r] = 3.4e38f; mini[r] = 0u; }

    for (int chunk = 0; chunk < NUM_CHUNKS; ++chunk) {
        const int base = chunk * CHUNK_COLS;
        // Cooperative stage: 256 threads x 8 independent b128 loads -> LDS
#pragma unroll
        for (int k = 0; k < CHUNK_COLS / 256; ++k) {
            sh[tid + k * 256] = gt4[base + tid + k * 256];
        }
        // Prefetch next chunk into cache while we compute (one 128B line per thread)
        if (chunk + 1 < NUM_CHUNKS) {
            __builtin_prefetch(&gt4[(chunk + 1) * CHUNK_COLS + tid * 8], 0, 1);
        }
        __syncthreads();

#pragma unroll 2
        for (int tile = 0; tile < CHUNK_TILES; ++tile) {
            const int jloc = tile * 16 + c16;
            const unsigned j = (unsigned)(base + jloc);    // this lane's global column
            float4 g = sh[jloc];                           // ds_load_b128
            v2f bm;
            bm.x = hiHalf ? (-2.0f * g.z) : (-2.0f * g.x);
            bm.y = hiHalf ? 1.0f          : (-2.0f * g.y);
            v8f cz = {};
            v8f d = __builtin_amdgcn_wmma_f32_16x16x4_f32(
                /*neg_a=*/false, a, /*neg_b=*/false, bm,
                /*c_mod=*/(short)0, cz, /*reuse_a=*/false, /*reuse_b=*/false);
            const float gn = g.w;
#pragma unroll
            for (int r = 0; r < 8; ++r) {
                float sq = fmaxf(d[r] + gn, 0.0f);   // matches sqrt(max(sq,0)) argmin ordering
                if (sq < minv[r]) { minv[r] = sq; mini[r] = j; }  // strict < keeps first occurrence
            }
        }
        __syncthreads();
    }

    // Reduce over the 16 columns held by each 16-lane half (masks 1..8 stay in-half).
#pragma unroll
    for (int r = 0; r < 8; ++r) {
        float v = minv[r];
        unsigned id = mini[r];
        for (int off = 1; off < 16; off <<= 1) {
            float    ov = __shfl_xor(v, off, 32);
            unsigned oi = (unsigned)__shfl_xor((int)id, off, 32);
            if (ov < v || (ov == v && oi < id)) { v = ov; id = oi; }
        }
        minv[r] = v; mini[r] = id;
    }

    // Lanes 0 and 16 own rows rowbase+r and rowbase+8+r respectively: balanced-BCE accumulation.
    if (c16 == 0) {
        float s1 = 0.f, c1 = 0.f, s0 = 0.f, c0 = 0.f;
#pragma unroll
        for (int r = 0; r < 8; ++r) {
            int i = rowbase + r + (hiHalf ? 8 : 0);
            float x = logits[i];
            float t = log1pf(expf(-fabsf(x)));        // stable softplus core
            float sp_pos = t + fmaxf(x, 0.0f);        // softplus(x)
            float sp_neg = t + fmaxf(-x, 0.0f);       // softplus(-x)
            if (mini[r] == (unsigned)i) { s1 += sp_neg; c1 += 1.0f; }
            else                        { s0 += sp_pos; c0 += 1.0f; }
        }
        atomicAdd(&acc[2], s1);
        atomicAdd(&acc[3], c1);
        atomicAdd(&acc[4], s0);
        atomicAdd(&acc[5], c0);
    }
}

__global__ void cc_final_kernel(const float* __restrict__ acc, float* __restrict__ out) {
    float mean_err = acc[0] / fmaxf(acc[1], 1e-6f);
    float g1 = (acc[3] > 0.0f) ? acc[2] / fmaxf(acc[3], 1.0f) : 0.0f;
    float g0 = (acc[5] > 0.0f) ? acc[4] / fmaxf(acc[5], 1.0f) : 0.0f;
    out[0] = mean_err + 0.5f * (g0 + g1);
}

extern "C" void kernel_launch(void* const* d_in, const int* in_sizes, int n_in,
                              void* d_out, int out_size, void* d_ws, size_t ws_size,
                              hipStream_t stream) {
    (void)in_sizes; (void)n_in; (void)out_size; (void)ws_size;
    const float* kp_before = (const float*)d_in[0];
    const float* pred      = (const float*)d_in[1];   // kp_warped_pred
    const float* pose      = (const float*)d_in[2];
    const float* w         = (const float*)d_in[3];
    const float* logits    = (const float*)d_in[4];

    float4* gt4 = (float4*)d_ws;                                    // M * 16 bytes
    float*  acc = (float*)((char*)d_ws + M_TOTAL * sizeof(float4)); // 6 floats

    cc_init_kernel<<<1, 32, 0, stream>>>(acc);
    cc_prep_kernel<<<M_TOTAL / 256, 256, 0, stream>>>(kp_before, pred, pose, w, gt4, acc);
    // 1024 row tiles, 8 waves (256 threads) per block -> 128 blocks
    cc_argmin_kernel<<<(M_TOTAL / 16) / 8, 256, 0, stream>>>(pred, gt4, logits, acc);
    cc_final_kernel<<<1, 1, 0, stream>>>(acc, (float*)d_out);
}